// OptimalTransportLayer_50319836840105
// MI455X (gfx1250) — compile-verified
//
#include <hip/hip_runtime.h>
#include <hip/hip_bf16.h>

#define OT_GAMMA 1.0f
#define OT_EPS   1e-6f
#define OT_MAXIT 1000
#define OT_B 64
#define OT_H 512
#define OT_W 512
#define NT  256   // 8 waves of 32

typedef __attribute__((ext_vector_type(2))) float v2f;
typedef __attribute__((ext_vector_type(8))) float v8f;

// ctrl layout in d_ws: [0]=arrive count, [1]=release generation, [2]=done flag, [3]=maxdiff bits (uint)
__global__ void ot_init_ctrl(int* ctrl) {
    ctrl[0] = 0; ctrl[1] = 0; ctrl[2] = 0; ctrl[3] = 0;
}

__global__ __launch_bounds__(NT) void ot_sinkhorn(
    const float* __restrict__ M, const float* __restrict__ r,
    const float* __restrict__ c, float* __restrict__ P, int* ctrl) {

    __shared__ float s_r[OT_H], s_c[OT_W], s_u[OT_H], s_v[OT_W];
    __shared__ float s_red[NT];
    __shared__ int   s_done;

    const int b    = blockIdx.x;
    const int t    = threadIdx.x;
    const int wave = t >> 5;
    const int lane = t & 31;
    const int half = lane >> 4;   // 0: K,K+1 lanes ; 1: K+2,K+3 lanes
    const int lrow = lane & 15;   // A-matrix row within tile

    const float* Mb = M + (size_t)b * OT_H * OT_W;
    float*       K  = P + (size_t)b * OT_H * OT_W;   // K lives in d_out until final scale

    // ---------- normalize r, c ; init u = v = 1 ----------
    {
        float r0 = r[b * OT_H + t], r1 = r[b * OT_H + t + 256];
        s_red[t] = r0 + r1;
        __syncthreads();
        for (int s = 128; s > 0; s >>= 1) { if (t < s) s_red[t] += s_red[t + s]; __syncthreads(); }
        float rsum = s_red[0];
        __syncthreads();
        s_r[t] = r0 / rsum; s_r[t + 256] = r1 / rsum;

        float c0 = c[b * OT_W + t], c1 = c[b * OT_W + t + 256];
        s_red[t] = c0 + c1;
        __syncthreads();
        for (int s = 128; s > 0; s >>= 1) { if (t < s) s_red[t] += s_red[t + s]; __syncthreads(); }
        float csum = s_red[0];
        __syncthreads();
        s_c[t] = c0 / csum; s_c[t + 256] = c1 / csum;

        s_u[t] = 1.0f; s_u[t + 256] = 1.0f;
        s_v[t] = 1.0f; s_v[t + 256] = 1.0f;
    }

    // ---------- K = exp(-gamma * (M - rowmin)) ----------
    for (int i = wave; i < OT_H; i += 8) {
        const float* Mrow = Mb + (size_t)i * OT_W;
        float mn = __builtin_inff();
        #pragma unroll 4
        for (int j = lane; j < OT_W; j += 32) mn = fminf(mn, Mrow[j]);
        #pragma unroll
        for (int off = 16; off > 0; off >>= 1) mn = fminf(mn, __shfl_xor(mn, off, 32));
        float* Krow = K + (size_t)i * OT_W;
        #pragma unroll 4
        for (int j = lane; j < OT_W; j += 32) Krow[j] = expf(-OT_GAMMA * (Mrow[j] - mn));
    }
    __syncthreads();

    // ---------- Sinkhorn loop ----------
    for (int it = 0; it < OT_MAXIT; ++it) {
        // ---- pass A: y = K v via V_WMMA_F32_16X16X4_F32 ; u <- u*r/alpha ----
        // wave handles 16-row groups g, g+8, g+16, g+24
        for (int g = wave; g < 32; g += 8) {
            const int rbase = g * 16;
            const float* Krow = K + (size_t)(rbase + lrow) * OT_W;
            v8f acc0 = {}, acc1 = {};
            #pragma unroll 4
            for (int k = 0; k < OT_W; k += 8) {
                int col0 = k + half * 2;
                int col1 = k + 4 + half * 2;
                v2f a0 = *(const v2f*)(Krow + col0);
                v2f a1 = *(const v2f*)(Krow + col1);
                v2f b0 = *(const v2f*)(s_v + col0);   // broadcast across N
                v2f b1 = *(const v2f*)(s_v + col1);
                acc0 = __builtin_amdgcn_wmma_f32_16x16x4_f32(
                    false, a0, false, b0, (short)0, acc0, false, false);
                acc1 = __builtin_amdgcn_wmma_f32_16x16x4_f32(
                    false, a1, false, b1, (short)0, acc1, false, false);
            }
            v8f acc = acc0 + acc1;
            // C layout: VGPR gg = row rbase+gg (lanes 0-15) / rbase+gg+8 (lanes 16-31), all N equal
            if (lrow == 0) {
                #pragma unroll
                for (int gg = 0; gg < 8; ++gg) {
                    int   row   = rbase + gg + half * 8;
                    float y     = acc[gg];
                    float uo    = s_u[row];
                    float alpha = uo * y;
                    if (alpha == 0.0f) alpha = OT_EPS;
                    s_u[row] = uo * s_r[row] / alpha;
                }
            }
        }
        __syncthreads();

        // ---- pass B: z = K^T u ; beta = v*z ; maxdiff ----
        const int j0 = 2 * t;
        float z0 = 0.0f, z1 = 0.0f;
        #pragma unroll 8
        for (int i = 0; i < OT_H; ++i) {
            float ui = s_u[i];
            v2f kk = *(const v2f*)(K + (size_t)i * OT_W + j0);
            z0 = fmaf(ui, kk.x, z0);
            z1 = fmaf(ui, kk.y, z1);
        }
        float v0 = s_v[j0], v1 = s_v[j0 + 1];
        float b0 = v0 * z0; if (b0 == 0.0f) b0 = OT_EPS;
        float b1 = v1 * z1; if (b1 == 0.0f) b1 = OT_EPS;
        float dm = fmaxf(fabsf(b0 - s_c[j0]), fabsf(b1 - s_c[j0 + 1]));

        s_red[t] = dm;
        __syncthreads();
        for (int s = 128; s > 0; s >>= 1) { if (t < s) s_red[t] = fmaxf(s_red[t], s_red[t + s]); __syncthreads(); }

        // ---- device-wide barrier + global done consensus ----
        if (t == 0) {
            unsigned bits = __float_as_uint(s_red[0]);
            __hip_atomic_fetch_max((unsigned*)&ctrl[3], bits,
                                   __ATOMIC_RELAXED, __HIP_MEMORY_SCOPE_AGENT);
            const int gen = it + 1;
            int prev = __hip_atomic_fetch_add(&ctrl[0], 1,
                                              __ATOMIC_ACQ_REL, __HIP_MEMORY_SCOPE_AGENT);
            if (prev == OT_B - 1) {
                unsigned mb = __hip_atomic_load((unsigned*)&ctrl[3],
                                                __ATOMIC_RELAXED, __HIP_MEMORY_SCOPE_AGENT);
                int dn = (__uint_as_float(mb) <= OT_EPS) ? 1 : 0;
                __hip_atomic_store((unsigned*)&ctrl[3], 0u,
                                   __ATOMIC_RELAXED, __HIP_MEMORY_SCOPE_AGENT);
                __hip_atomic_store(&ctrl[2], dn,
                                   __ATOMIC_RELAXED, __HIP_MEMORY_SCOPE_AGENT);
                __hip_atomic_store(&ctrl[0], 0,
                                   __ATOMIC_RELAXED, __HIP_MEMORY_SCOPE_AGENT);
                __hip_atomic_store(&ctrl[1], gen,
                                   __ATOMIC_RELEASE, __HIP_MEMORY_SCOPE_AGENT);
                s_done = dn;
            } else {
                while (__hip_atomic_load(&ctrl[1], __ATOMIC_ACQUIRE,
                                         __HIP_MEMORY_SCOPE_AGENT) < gen) {
                    __builtin_amdgcn_s_sleep(2);
                }
                s_done = __hip_atomic_load(&ctrl[2], __ATOMIC_RELAXED,
                                           __HIP_MEMORY_SCOPE_AGENT);
            }
        }
        __syncthreads();
        if (s_done) break;                 // converged: skip column update (matches reference)

        // ---- column update: v <- v*c/beta ----
        s_v[j0]     = v0 * s_c[j0]     / b0;
        s_v[j0 + 1] = v1 * s_c[j0 + 1] / b1;
        __syncthreads();
    }

    // ---------- final: P = diag(u) K diag(v), in place in d_out ----------
    float4* K4 = (float4*)K;
    #pragma unroll 4
    for (int idx = t; idx < (OT_H * OT_W) / 4; idx += NT) {
        int i = idx >> 7;           // W/4 == 128 float4 per row
        int j = (idx & 127) << 2;
        float4 kk = K4[idx];
        float ui = s_u[i];
        kk.x *= ui * s_v[j];
        kk.y *= ui * s_v[j + 1];
        kk.z *= ui * s_v[j + 2];
        kk.w *= ui * s_v[j + 3];
        K4[idx] = kk;
    }
}

extern "C" void kernel_launch(void* const* d_in, const int* in_sizes, int n_in,
                              void* d_out, int out_size, void* d_ws, size_t ws_size,
                              hipStream_t stream) {
    const float* M = (const float*)d_in[0];
    const float* r = (const float*)d_in[1];
    const float* c = (const float*)d_in[2];
    float* P = (float*)d_out;
    int* ctrl = (int*)d_ws;

    ot_init_ctrl<<<1, 1, 0, stream>>>(ctrl);
    ot_sinkhorn<<<OT_B, NT, 0, stream>>>(M, r, c, P, ctrl);
}